// GCNWithPooling_73787538145377
// MI455X (gfx1250) — compile-verified
//
#include <hip/hip_runtime.h>

#define NN 65536
#define EE 1048576
#define HH 64
#define CC 10
#define GG 256
#define LL 3
#define BN_EPS 1e-5f

typedef __attribute__((ext_vector_type(2))) float v2f;
typedef __attribute__((ext_vector_type(8))) float v8f;

// ---------------- degree / normalization ----------------
__global__ void k_init_deg(float* dis) {
    int n = blockIdx.x * 256 + threadIdx.x;
    dis[n] = 1.0f;                      // self-loop counted once
}

__global__ void k_count_deg(const int* __restrict__ dst, float* dis) {
    int e = blockIdx.x * 256 + threadIdx.x;
    atomicAdd(&dis[dst[e]], 1.0f);
}

__global__ void k_finish_deg(float* dis) {
    int n = blockIdx.x * 256 + threadIdx.x;
    dis[n] = rsqrtf(dis[n]);
}

// ---------------- fp32 WMMA GEMM: Y[nrows,64] = X[nrows,64] @ W[64,64] ----------------
// One wave computes a 16-row x 64-col tile using V_WMMA_F32_16X16X4_F32.
// Layouts per CDNA5 ISA 7.12.2:
//  A 16x4 : lanes 0-15 hold K={0,1} in v0/v1, lanes 16-31 hold K={2,3}
//  B 4x16 : v0 holds rows {0,2} (lane halves), v1 holds rows {1,3}
//  C/D    : VGPR j holds row j (lanes 0-15) and row j+8 (lanes 16-31)
// Epilogue additionally writes agg = Y * dis[row]^2 (GCN self-loop term), which
// removes a separate 32MB init pass per layer.
__global__ void k_gemm64(const float* __restrict__ X, const float* __restrict__ W,
                         const float* __restrict__ dis,
                         float* __restrict__ Y, float* __restrict__ agg) {
    int wave = (int)((blockIdx.x * (unsigned)blockDim.x + threadIdx.x) >> 5);
    int lane = threadIdx.x & 31;
    int half = lane >> 4;               // 0 or 1
    int l15  = lane & 15;
    int row  = wave * 16 + l15;
    int kh   = half * 2;                // K sub-offset for this lane half

    v8f acc0 = {}, acc1 = {}, acc2 = {}, acc3 = {};

#pragma unroll
    for (int k = 0; k < 64; k += 4) {
        v2f a;
        a[0] = X[(size_t)row * 64 + k + kh];
        a[1] = X[(size_t)row * 64 + k + kh + 1];
        const float* Wk = W + (size_t)(k + kh) * 64;   // row (k+kh) of W
        v2f b0, b1, b2, b3;
        b0[0] = Wk[ 0 + l15];  b0[1] = Wk[64 +  0 + l15];
        b1[0] = Wk[16 + l15];  b1[1] = Wk[64 + 16 + l15];
        b2[0] = Wk[32 + l15];  b2[1] = Wk[64 + 32 + l15];
        b3[0] = Wk[48 + l15];  b3[1] = Wk[64 + 48 + l15];
        acc0 = __builtin_amdgcn_wmma_f32_16x16x4_f32(false, a, false, b0, (short)0, acc0, false, false);
        acc1 = __builtin_amdgcn_wmma_f32_16x16x4_f32(false, a, false, b1, (short)0, acc1, false, false);
        acc2 = __builtin_amdgcn_wmma_f32_16x16x4_f32(false, a, false, b2, (short)0, acc2, false, false);
        acc3 = __builtin_amdgcn_wmma_f32_16x16x4_f32(false, a, false, b3, (short)0, acc3, false, false);
    }

    int r0 = wave * 16 + half * 8;
#pragma unroll
    for (int j = 0; j < 8; ++j) {
        int r = r0 + j;
        float d  = dis[r];
        float d2 = d * d;
        size_t base = (size_t)r * 64 + l15;
        Y[base +  0] = acc0[j];
        Y[base + 16] = acc1[j];
        Y[base + 32] = acc2[j];
        Y[base + 48] = acc3[j];
        agg[base +  0] = acc0[j] * d2;
        agg[base + 16] = acc1[j] * d2;
        agg[base + 32] = acc2[j] * d2;
        agg[base + 48] = acc3[j] * d2;
    }
}

// ---------------- edge aggregation ----------------
// 16 lanes per edge, float4 per lane: b128 coalesced gather of the 256B row,
// then 4 L2-resident f32 atomics. This is the bandwidth-dominant kernel.
__global__ void k_edge_agg(const float* __restrict__ hw, const float* __restrict__ dis,
                           const int* __restrict__ src, const int* __restrict__ dst,
                           float* __restrict__ agg) {
    long long t = (long long)blockIdx.x * 256 + threadIdx.x;  // EE*16 threads
    int e  = (int)(t >> 4);
    int f4 = (int)(t & 15) * 4;
    int s = src[e];
    int d = dst[e];
    float c = dis[s] * dis[d];
    const float4 v = *(const float4*)(hw + (size_t)s * 64 + f4);
    float* ap = agg + (size_t)d * 64 + f4;
    atomicAdd(ap + 0, v.x * c);
    atomicAdd(ap + 1, v.y * c);
    atomicAdd(ap + 2, v.z * c);
    atomicAdd(ap + 3, v.w * c);
}

// bias + BatchNorm(eval) + ReLU + optional residual, float4 per thread, h in place
__global__ void k_post(const float* __restrict__ agg, const float* __restrict__ bias,
                       const float* __restrict__ gamma, const float* __restrict__ beta,
                       const float* __restrict__ mean, const float* __restrict__ var,
                       float* __restrict__ h, int add_res) {
    size_t t = (size_t)blockIdx.x * 256 + threadIdx.x;   // NN*16 threads
    int f4 = (int)(t & 15) * 4;
    size_t base = (t >> 4) * 64 + f4;
    float4 a = *(const float4*)(agg + base);
    float4 r;
    float va[4] = {a.x, a.y, a.z, a.w};
    float vo[4];
#pragma unroll
    for (int j = 0; j < 4; ++j) {
        int f = f4 + j;
        float v = va[j] + bias[f];
        v = gamma[f] * (v - mean[f]) * rsqrtf(var[f] + BN_EPS) + beta[f];
        vo[j] = fmaxf(v, 0.0f);
    }
    if (add_res) {
        float4 hv = *(const float4*)(h + base);
        vo[0] += hv.x; vo[1] += hv.y; vo[2] += hv.z; vo[3] += hv.w;
    }
    r.x = vo[0]; r.y = vo[1]; r.z = vo[2]; r.w = vo[3];
    *(float4*)(h + base) = r;
}

// ---------------- pooling ----------------
__global__ void k_zero(float* p, int n) {
    int i = blockIdx.x * 256 + threadIdx.x;
    if (i < n) p[i] = 0.0f;
}

__global__ void k_pool(const float* __restrict__ h, const int* __restrict__ batch,
                       float* __restrict__ sums, float* __restrict__ counts) {
    size_t t = (size_t)blockIdx.x * 256 + threadIdx.x;   // NN*16 threads
    int n  = (int)(t >> 4);
    int f4 = (int)(t & 15) * 4;
    int g = batch[n];
    float4 v = *(const float4*)(h + (size_t)n * 64 + f4);
    float* sp = sums + (size_t)g * 64 + f4;
    atomicAdd(sp + 0, v.x);
    atomicAdd(sp + 1, v.y);
    atomicAdd(sp + 2, v.z);
    atomicAdd(sp + 3, v.w);
    if (f4 == 0) atomicAdd(&counts[g], 1.0f);
}

// ---------------- classifier head: mean -> Linear+ReLU -> Linear -> log_softmax ----------------
__global__ void k_cls(const float* __restrict__ sums, const float* __restrict__ counts,
                      const float* __restrict__ W1, const float* __restrict__ b1,
                      const float* __restrict__ W2, const float* __restrict__ b2,
                      float* __restrict__ out) {
    __shared__ float gr[64];
    __shared__ float z[64];
    __shared__ float logit[16];
    int g = blockIdx.x;
    int f = threadIdx.x;                 // 64 threads (2 waves)
    float cnt = fmaxf(counts[g], 1.0f);
    gr[f] = sums[(size_t)g * 64 + f] / cnt;
    __syncthreads();
    float acc = b1[f];
#pragma unroll 8
    for (int k = 0; k < 64; ++k) acc += gr[k] * W1[k * 64 + f];
    z[f] = fmaxf(acc, 0.0f);
    __syncthreads();
    if (f < CC) {
        float a2 = b2[f];
#pragma unroll 8
        for (int k = 0; k < 64; ++k) a2 += z[k] * W2[k * CC + f];
        logit[f] = a2;
    }
    __syncthreads();
    if (f == 0) {
        float m = logit[0];
        for (int c = 1; c < CC; ++c) m = fmaxf(m, logit[c]);
        float s = 0.0f;
        for (int c = 0; c < CC; ++c) s += __expf(logit[c] - m);
        float lse = m + __logf(s);
        for (int c = 0; c < CC; ++c) out[(size_t)g * CC + c] = logit[c] - lse;
    }
}

extern "C" void kernel_launch(void* const* d_in, const int* in_sizes, int n_in,
                              void* d_out, int out_size, void* d_ws, size_t ws_size,
                              hipStream_t stream) {
    (void)in_sizes; (void)n_in; (void)out_size; (void)ws_size;
    const float* x      = (const float*)d_in[0];
    const float* conv_w = (const float*)d_in[1];   // [L,64,64]
    const float* conv_b = (const float*)d_in[2];   // [L,64]
    const float* bn_g   = (const float*)d_in[3];
    const float* bn_b   = (const float*)d_in[4];
    const float* bn_m   = (const float*)d_in[5];
    const float* bn_v   = (const float*)d_in[6];
    const float* w1     = (const float*)d_in[7];   // [64,64]
    const float* b1     = (const float*)d_in[8];
    const float* w2     = (const float*)d_in[9];   // [64,10]
    const float* b2     = (const float*)d_in[10];
    const int*   eidx   = (const int*)d_in[11];    // [2,E]
    const int*   batch  = (const int*)d_in[12];    // [N]
    const int* src = eidx;
    const int* dst = eidx + EE;
    float* out = (float*)d_out;

    // workspace layout (fp32): dis[N] | hw[N*64] | agg[N*64] | h[N*64] | sums[G*64] | counts[G]
    float* dis    = (float*)d_ws;
    float* hw     = dis + NN;
    float* agg    = hw + (size_t)NN * HH;
    float* h      = agg + (size_t)NN * HH;
    float* sums   = h + (size_t)NN * HH;
    float* counts = sums + (size_t)GG * HH;

    const unsigned NF4_BLOCKS = (unsigned)((size_t)NN * 16 / 256);       // 4096
    const unsigned EF4_BLOCKS = (unsigned)((size_t)EE * 16 / 256);       // 65536
    const unsigned GEMM_BLOCKS = (unsigned)((NN / 16) * 32 / 256);       // 512

    // degree + symmetric normalization
    k_init_deg  <<<NN / 256, 256, 0, stream>>>(dis);
    k_count_deg <<<EE / 256, 256, 0, stream>>>(dst, dis);
    k_finish_deg<<<NN / 256, 256, 0, stream>>>(dis);

    for (int l = 0; l < LL; ++l) {
        const float* Xin = (l == 0) ? x : h;
        k_gemm64  <<<GEMM_BLOCKS, 256, 0, stream>>>(Xin, conv_w + (size_t)l * HH * HH, dis, hw, agg);
        k_edge_agg<<<EF4_BLOCKS, 256, 0, stream>>>(hw, dis, src, dst, agg);
        k_post    <<<NF4_BLOCKS, 256, 0, stream>>>(agg, conv_b + (size_t)l * HH,
                                                   bn_g + (size_t)l * HH, bn_b + (size_t)l * HH,
                                                   bn_m + (size_t)l * HH, bn_v + (size_t)l * HH,
                                                   h, l > 0 ? 1 : 0);
    }

    k_zero<<<(GG * HH + GG + 255) / 256, 256, 0, stream>>>(sums, GG * HH + GG);  // sums+counts contiguous
    k_pool<<<NF4_BLOCKS, 256, 0, stream>>>(h, batch, sums, counts);
    k_cls <<<GG, 64, 0, stream>>>(sums, counts, w1, b1, w2, b2, out);
}